// GraphRNN_82755429860278
// MI455X (gfx1250) — compile-verified
//
#include <hip/hip_runtime.h>
#include <cstdint>

typedef float v2f __attribute__((ext_vector_type(2)));
typedef float v8f __attribute__((ext_vector_type(8)));

__device__ __forceinline__ float sigf(float v) { return 1.0f / (1.0f + expf(-v)); }

// ---------------- utility kernels ----------------
__global__ void k_fill(float* __restrict__ p, float v, long long n) {
  long long i = (long long)blockIdx.x * blockDim.x + threadIdx.x;
  if (i < n) p[i] = v;
}

__global__ void k_deg(const int* __restrict__ dst, const float* __restrict__ ew,
                      float* __restrict__ deg, int E) {
  int e = blockIdx.x * blockDim.x + threadIdx.x;
  if (e < E) atomicAdd(&deg[dst[e]], ew[e]);
}

__global__ void k_rsqrt(float* __restrict__ d, int n) {
  int i = blockIdx.x * blockDim.x + threadIdx.x;
  if (i < n) { float v = d[i]; d[i] = (v > 0.0f) ? rsqrtf(v) : 0.0f; }
}

__global__ void k_norm(const int* __restrict__ src, const int* __restrict__ dst,
                       const float* __restrict__ ew, const float* __restrict__ dinv,
                       float* __restrict__ nrm, int E) {
  int e = blockIdx.x * blockDim.x + threadIdx.x;
  if (e < E) nrm[e] = dinv[src[e]] * ew[e] * dinv[dst[e]];
}

// s[i] = dinv[i]^2 * x[i]   (self-loop term of the scalar layer-0 scatter)
__global__ void k_s_init(const float* __restrict__ x, const float* __restrict__ dinv,
                         float* __restrict__ s, int n) {
  int i = blockIdx.x * blockDim.x + threadIdx.x;
  if (i < n) { float di = dinv[i]; s[i] = di * di * x[i]; }
}

__global__ void k_s_scatter(const int* __restrict__ src, const int* __restrict__ dst,
                            const float* __restrict__ nrm, const float* __restrict__ x,
                            float* __restrict__ s, int E) {
  int e = blockIdx.x * blockDim.x + threadIdx.x;
  if (e < E) atomicAdd(&s[dst[e]], nrm[e] * x[src[e]]);
}

// x1[n,j] = relu(s[n]*W0[j] + b0[j])   (T==1: layer-0 GEMM is an outer product)
__global__ void k_layer0(const float* __restrict__ s, const float* __restrict__ W0,
                         const float* __restrict__ b0, float* __restrict__ xo, int NT) {
  long long t = (long long)blockIdx.x * blockDim.x + threadIdx.x;
  int n = (int)(t >> 6);
  if (n >= NT) return;
  int j = ((int)t & 63) << 2;
  float sv = s[n];
  float4 w = *(const float4*)(W0 + j);
  float4 b = *(const float4*)(b0 + j);
  float4 o;
  o.x = fmaxf(fmaf(sv, w.x, b.x), 0.0f);
  o.y = fmaxf(fmaf(sv, w.y, b.y), 0.0f);
  o.z = fmaxf(fmaf(sv, w.z, b.z), 0.0f);
  o.w = fmaxf(fmaf(sv, w.w, b.w), 0.0f);
  *(float4*)(xo + (size_t)n * 256 + j) = o;
}

// ---------------- B-matrix fragment packing ----------------
// Packed layout: Bp[(p*N + n)*2 + {0,1}] = (B[2p][n], B[2p+1][n]); p = K-pair index.
// A lane's WMMA B-fragment (K rows k+2*hi, k+2*hi+1 at column n) is then one b64 load.

// from row-major B[Kdim x N]
__global__ void k_pack_b(const float* __restrict__ B, float* __restrict__ Bp,
                         int Kdim, int N) {
  long long t = (long long)blockIdx.x * blockDim.x + threadIdx.x;
  long long tot = (long long)(Kdim / 2) * N;
  if (t >= tot) return;
  int p = (int)(t / N);
  int n = (int)(t - (long long)p * N);
  float2 v;
  v.x = B[(size_t)(2 * p) * N + n];
  v.y = B[(size_t)(2 * p + 1) * N + n];
  *(float2*)(Bp + ((size_t)p * N + n) * 2) = v;
}

// from W[N x Kdim] row-major, packing B = W^T (fuses the transpose: pair is contiguous in W)
__global__ void k_pack_bT(const float* __restrict__ W, float* __restrict__ Bp,
                          int Kdim, int N) {
  long long t = (long long)blockIdx.x * blockDim.x + threadIdx.x;
  long long tot = (long long)(Kdim / 2) * N;
  if (t >= tot) return;
  int p = (int)(t / N);
  int n = (int)(t - (long long)p * N);
  float2 v = *(const float2*)(W + (size_t)n * Kdim + 2 * p);
  *(float2*)(Bp + ((size_t)p * N + n) * 2) = v;
}

// ---------------- fp32 WMMA GEMM:  C[M x Ncols] (+)= A[M x 256] @ B[256 x ...] ----------------
// B is pair-packed (see above) with packed row width nB (>= Ncols; base pointer may be
// column-offset by colInB*2). One wave computes a 16x64 tile via V_WMMA_F32_16X16X4_F32.
// Inner loop: 1 global b64 (A) + 4 global b64 (B) per 4 WMMAs.
__global__ __launch_bounds__(256) void k_gemm_wmma(
    const float* __restrict__ A, const float* __restrict__ Bp, float* __restrict__ C,
    int M, int Ncols, int nB, int ldc, int accum) {
  const int K = 256;
  int wid = blockIdx.x * (blockDim.x >> 5) + (threadIdx.x >> 5);
  int lane = threadIdx.x & 31;
  int tilesN = Ncols >> 6;
  int rt = wid / tilesN;
  int row0 = rt << 4;
  if (row0 >= M) return;                       // whole-wave exit: EXEC all-1s at WMMA
  int col0 = (wid - rt * tilesN) << 6;
  int m  = lane & 15;
  int hi = lane >> 4;                          // lanes 16-31 hold K+2/K+3 halves

  const float* ap = A + (size_t)(row0 + m) * K + 2 * hi;          // A[m][k+2*hi .. +1]
  const float* bp = Bp + ((size_t)hi * nB + col0 + m) * 2;        // pair row k/2+hi, col

  v8f acc0 = {}, acc1 = {}, acc2 = {}, acc3 = {};
  for (int k = 0; k < K; k += 4) {
    v2f a = *(const v2f*)(ap + k);
    const float* b = bp + (size_t)k * nB;      // (k/2 pairs) * nB * 2 elems
    v2f b0 = *(const v2f*)(b);
    v2f b1 = *(const v2f*)(b + 32);            // +16 columns
    v2f b2 = *(const v2f*)(b + 64);
    v2f b3 = *(const v2f*)(b + 96);
    acc0 = __builtin_amdgcn_wmma_f32_16x16x4_f32(false, a, false, b0, (short)0, acc0, false, false);
    acc1 = __builtin_amdgcn_wmma_f32_16x16x4_f32(false, a, false, b1, (short)0, acc1, false, false);
    acc2 = __builtin_amdgcn_wmma_f32_16x16x4_f32(false, a, false, b2, (short)0, acc2, false, false);
    acc3 = __builtin_amdgcn_wmma_f32_16x16x4_f32(false, a, false, b3, (short)0, acc3, false, false);
  }
  // D layout: element r of accumulator is row M = r + 8*hi, col N = lane&15 (+16*tile)
  float* crow = C + (size_t)(row0 + 8 * hi) * ldc + col0 + m;
  if (accum) {
    for (int r = 0; r < 8; ++r) {
      float* cp = crow + (size_t)r * ldc;
      cp[0] += acc0[r]; cp[16] += acc1[r]; cp[32] += acc2[r]; cp[48] += acc3[r];
    }
  } else {
    for (int r = 0; r < 8; ++r) {
      float* cp = crow + (size_t)r * ldc;
      cp[0] = acc0[r]; cp[16] = acc1[r]; cp[32] = acc2[r]; cp[48] = acc3[r];
    }
  }
}

// agg[dst,:] += norm_e * y[src,:]   (64 threads/edge, float4 each; coalesced row reads)
__global__ void k_scatter_vec(const int* __restrict__ src, const int* __restrict__ dst,
                              const float* __restrict__ nrm, const float* __restrict__ y,
                              float* __restrict__ agg, int E) {
  long long t = (long long)blockIdx.x * blockDim.x + threadIdx.x;
  int e = (int)(t >> 6);
  if (e >= E) return;
  int c = ((int)t & 63) << 2;
  int s = src[e], d = dst[e];
  float w = nrm[e];
  float4 v = *(const float4*)(y + (size_t)s * 256 + c);
  float* o = agg + (size_t)d * 256 + c;
  atomicAdd(o + 0, w * v.x);
  atomicAdd(o + 1, w * v.y);
  atomicAdd(o + 2, w * v.z);
  atomicAdd(o + 3, w * v.w);
}

// xnext = relu(agg + dinv^2 * y + b)   (self-loop folded in here)
__global__ void k_epilogue(const float* __restrict__ agg, const float* __restrict__ y,
                           const float* __restrict__ dinv, const float* __restrict__ b,
                           float* __restrict__ xo, int NT) {
  long long t = (long long)blockIdx.x * blockDim.x + threadIdx.x;
  int n = (int)(t >> 6);
  if (n >= NT) return;
  int j = ((int)t & 63) << 2;
  float di = dinv[n]; float c2 = di * di;
  float4 a = *(const float4*)(agg + (size_t)n * 256 + j);
  float4 v = *(const float4*)(y + (size_t)n * 256 + j);
  float4 bb = *(const float4*)(b + j);
  float4 o;
  o.x = fmaxf(a.x + c2 * v.x + bb.x, 0.0f);
  o.y = fmaxf(a.y + c2 * v.y + bb.y, 0.0f);
  o.z = fmaxf(a.z + c2 * v.z + bb.z, 0.0f);
  o.w = fmaxf(a.w + c2 * v.w + bb.w, 0.0f);
  *(float4*)(xo + (size_t)n * 256 + j) = o;
}

// GRU gates. gsum[NT x 768]: cols 0..511 = gx+gh (r,z), cols 512..767 = gx_n only.
// ghn[NT x 256] = h @ W_hh_n^T.
__global__ void k_gates(const float* __restrict__ gsum, const float* __restrict__ ghn,
                        const float* __restrict__ bih, const float* __restrict__ bhh,
                        const float* __restrict__ h, float* __restrict__ hn, int NT) {
  long long t = (long long)blockIdx.x * blockDim.x + threadIdx.x;
  int n = (int)(t >> 6);
  if (n >= NT) return;
  int j = ((int)t & 63) << 2;
  size_t g0 = (size_t)n * 768 + j;
  float4 gr = *(const float4*)(gsum + g0);
  float4 gz = *(const float4*)(gsum + g0 + 256);
  float4 gn = *(const float4*)(gsum + g0 + 512);
  float4 gh = *(const float4*)(ghn + (size_t)n * 256 + j);
  float4 hv = *(const float4*)(h + (size_t)n * 256 + j);
  float4 bri = *(const float4*)(bih + j);
  float4 bzi = *(const float4*)(bih + 256 + j);
  float4 bni = *(const float4*)(bih + 512 + j);
  float4 brh = *(const float4*)(bhh + j);
  float4 bzh = *(const float4*)(bhh + 256 + j);
  float4 bnh = *(const float4*)(bhh + 512 + j);
  float4 o;
  {
    float r = sigf(gr.x + bri.x + brh.x);
    float z = sigf(gz.x + bzi.x + bzh.x);
    float nn = tanhf(gn.x + bni.x + r * (gh.x + bnh.x));
    o.x = (1.0f - z) * nn + z * hv.x;
  }
  {
    float r = sigf(gr.y + bri.y + brh.y);
    float z = sigf(gz.y + bzi.y + bzh.y);
    float nn = tanhf(gn.y + bni.y + r * (gh.y + bnh.y));
    o.y = (1.0f - z) * nn + z * hv.y;
  }
  {
    float r = sigf(gr.z + bri.z + brh.z);
    float z = sigf(gz.z + bzi.z + bzh.z);
    float nn = tanhf(gn.z + bni.z + r * (gh.z + bnh.z));
    o.z = (1.0f - z) * nn + z * hv.z;
  }
  {
    float r = sigf(gr.w + bri.w + brh.w);
    float z = sigf(gz.w + bzi.w + bzh.w);
    float nn = tanhf(gn.w + bni.w + r * (gh.w + bnh.w));
    o.w = (1.0f - z) * nn + z * hv.w;
  }
  *(float4*)(hn + (size_t)n * 256 + j) = o;
}

// out[n] = hn[n,:] . Wfc + bfc   (one wave32 per node)
__global__ void k_fc(const float* __restrict__ hn, const float* __restrict__ wfc,
                     const float* __restrict__ bfc, float* __restrict__ out, int NT) {
  int wave = blockIdx.x * (blockDim.x >> 5) + (threadIdx.x >> 5);
  int lane = threadIdx.x & 31;
  if (wave >= NT) return;
  const float* row = hn + (size_t)wave * 256;
  float sum = 0.0f;
  for (int j = lane; j < 256; j += 32) sum += row[j] * wfc[j];
  for (int off = 16; off; off >>= 1) sum += __shfl_down(sum, off, 32);
  if (lane == 0) out[wave] = sum + bfc[0];
}

static inline unsigned nblk(long long n) { return (unsigned)((n + 255) / 256); }

extern "C" void kernel_launch(void* const* d_in, const int* in_sizes, int n_in,
                              void* d_out, int out_size, void* d_ws, size_t ws_size,
                              hipStream_t stream) {
  const float* x   = (const float*)d_in[0];
  const int*   ei  = (const int*)d_in[1];
  const float* ew  = (const float*)d_in[2];
  const float* h   = (const float*)d_in[3];
  const float* W0  = (const float*)d_in[4];
  const float* b0  = (const float*)d_in[5];
  const float* W1  = (const float*)d_in[6];
  const float* b1  = (const float*)d_in[7];
  const float* W2  = (const float*)d_in[8];
  const float* b2  = (const float*)d_in[9];
  const float* Wih = (const float*)d_in[10];
  const float* Whh = (const float*)d_in[11];
  const float* bih = (const float*)d_in[12];
  const float* bhh = (const float*)d_in[13];
  const float* Wfc = (const float*)d_in[14];
  const float* bfc = (const float*)d_in[15];

  const int NT = in_sizes[0];       // B*N_NODES*T, T==1
  const int E  = in_sizes[1] / 2;
  const int H = 256, K = 256, G3 = 768;
  const int* srcIdx = ei;
  const int* dstIdx = ei + E;

  float* out_scalar = (float*)d_out;          // [NT]
  float* hn = (float*)d_out + NT;             // [NT, H]

  // workspace layout
  float* p = (float*)d_ws;
  float* dinv = p; p += NT;
  float* sbuf = p; p += NT;
  float* nrm  = p; p += E;
  float* Bpk  = p; p += (size_t)K * G3;        // pair-packed B (reused per GEMM), max 256x768
  float* xb   = p; p += (size_t)NT * H;        // current features
  float* big  = p;                             // NT*1024 floats
  float* yb   = big;                           // GCN:  y   [NT x 256]
  float* ab   = big + (size_t)NT * H;          // GCN:  agg [NT x 256]
  float* gsum = big;                           // GRU:  [NT x 768]
  float* ghn  = big + (size_t)NT * G3;         // GRU:  gh_n [NT x 256]

  // ---- normalization ----
  k_fill<<<nblk(NT), 256, 0, stream>>>(dinv, 1.0f, (long long)NT);   // self-loop weight
  k_deg<<<nblk(E), 256, 0, stream>>>(dstIdx, ew, dinv, E);
  k_rsqrt<<<nblk(NT), 256, 0, stream>>>(dinv, NT);
  k_norm<<<nblk(E), 256, 0, stream>>>(srcIdx, dstIdx, ew, dinv, nrm, E);

  // ---- layer 0 (scalar scatter, then outer-product expand) ----
  k_s_init<<<nblk(NT), 256, 0, stream>>>(x, dinv, sbuf, NT);
  k_s_scatter<<<nblk(E), 256, 0, stream>>>(srcIdx, dstIdx, nrm, x, sbuf, E);
  k_layer0<<<nblk((long long)NT * 64), 256, 0, stream>>>(sbuf, W0, b0, xb, NT);

  // ---- layers 1..2: pack -> GEMM -> scatter -> epilogue ----
  const float* Ws[2] = {W1, W2};
  const float* bs[2] = {b1, b2};
  const long long packH = (long long)(K / 2) * H;
  for (int l = 0; l < 2; ++l) {
    k_pack_b<<<nblk(packH), 256, 0, stream>>>(Ws[l], Bpk, K, H);
    int waves = (NT / 16) * (H / 64);
    k_gemm_wmma<<<(waves + 7) / 8, 256, 0, stream>>>(xb, Bpk, yb, NT, H, H, H, 0);
    k_fill<<<nblk((long long)NT * H), 256, 0, stream>>>(ab, 0.0f, (long long)NT * H);
    k_scatter_vec<<<nblk((long long)E * 64), 256, 0, stream>>>(srcIdx, dstIdx, nrm, yb, ab, E);
    k_epilogue<<<nblk((long long)NT * 64), 256, 0, stream>>>(ab, yb, dinv, bs[l], xb, NT);
  }

  // ---- GRU cell (pack fuses the [3H,H] -> [H,3H] transpose) ----
  const long long packG = (long long)(K / 2) * G3;
  k_pack_bT<<<nblk(packG), 256, 0, stream>>>(Wih, Bpk, K, G3);
  {
    int waves = (NT / 16) * (G3 / 64);
    k_gemm_wmma<<<(waves + 7) / 8, 256, 0, stream>>>(xb, Bpk, gsum, NT, G3, G3, G3, 0);   // gx (all gates)
  }
  k_pack_bT<<<nblk(packG), 256, 0, stream>>>(Whh, Bpk, K, G3);
  {
    int waves_rz = (NT / 16) * (512 / 64);
    k_gemm_wmma<<<(waves_rz + 7) / 8, 256, 0, stream>>>(h, Bpk, gsum, NT, 512, G3, G3, 1); // += gh (r,z)
    int waves_n = (NT / 16) * (H / 64);
    k_gemm_wmma<<<(waves_n + 7) / 8, 256, 0, stream>>>(h, Bpk + 512 * 2, ghn, NT, H, G3, H, 0); // gh_n
  }
  k_gates<<<nblk((long long)NT * 64), 256, 0, stream>>>(gsum, ghn, bih, bhh, h, hn, NT);

  // ---- final linear ----
  k_fc<<<(NT + 7) / 8, 256, 0, stream>>>(hn, Wfc, bfc, out_scalar, NT);
}